// EmbeddingLayer_32392643347009
// MI455X (gfx1250) — compile-verified
//
#include <hip/hip_runtime.h>
#include <hip/hip_bf16.h>

// Problem constants (from reference setup_inputs)
#define B_   16
#define N_   2048
#define F_   64
#define E_   24576          // N*M edges per batch
#define C_   128            // 2F == bond == 128
#define CNT_ (B_ * E_)      // samples per channel for BN stats

typedef __attribute__((ext_vector_type(2))) float v2f;
typedef __attribute__((ext_vector_type(8))) float v8f;

// Workspace layout (floats)
#define WS_ABS   0          // 2048: per-(b,c) L1 abs sums
#define WS_RD    2048       // 2048: reciprocal denom
#define WS_SUM   4096       // 128 : per-channel sum(y)
#define WS_SQ    4224       // 128 : per-channel sum(y^2)
#define WS_SCALE 4352       // 128 : gamma * rstd
#define WS_SHIFT 4480       // 128 : beta - mean*scale
#define WS_FLOATS 4608

// WkP pair-row stride in floats: 288 % 64 == 32 -> the two K-pair rows used by
// the two half-waves land on disjoint bank halves (0-31 vs 32-63).
#define WKP_STRIDE 288

// ---------------------------------------------------------------------------
// Pass 1: L1 sums over the edge dimension. x[b,e,c] = atom[b, adj[b,e,c/64], c%64]
// atom (8 MB) is L2-resident; gather reads mostly hit L2.
// ---------------------------------------------------------------------------
__global__ __launch_bounds__(256) void k_l1sum(const float* __restrict__ atom,
                                               const int* __restrict__ adj,
                                               float* __restrict__ ws) {
  const int b  = blockIdx.y;
  const int e0 = blockIdx.x * 256;
  const int c  = threadIdx.x & 127;
  const int eh = threadIdx.x >> 7;        // 0/1: split edges over the two half-blocks
  const float* atomB = atom + (size_t)b * N_ * F_;
  const int*   adjB  = adj  + (size_t)b * E_ * 2;
  float acc = 0.f;
  for (int e = e0 + eh; e < e0 + 256; e += 2) {
    const int idx = adjB[e * 2 + (c >> 6)];
    acc += fabsf(atomB[idx * F_ + (c & 63)]);
  }
  atomicAdd(&ws[WS_ABS + b * C_ + c], acc);
}

// ---------------------------------------------------------------------------
// Pass 1b: rdenom = 1 / max(L1sum, 1e-12)
// ---------------------------------------------------------------------------
__global__ void k_rdenom(float* __restrict__ ws) {
  const int i = blockIdx.x * 256 + threadIdx.x;
  if (i < B_ * C_) ws[WS_RD + i] = 1.0f / fmaxf(ws[WS_ABS + i], 1e-12f);
}

// ---------------------------------------------------------------------------
// Pass 2: y = (x * rdenom) @ W + bias  via f32 WMMA, writing y to d_out and
// accumulating per-channel sum / sumsq for BatchNorm.
// Block: 256 threads (8 waves), tile = 128 edges x 128 channels, K = 128 in
// four K-slices of 32. rdenom is folded into the W rows (per batch).
// W slice is stored K-pair-interleaved so each B fragment is one ds_load_b64.
// ---------------------------------------------------------------------------
__global__ __launch_bounds__(256) void k_gemm(const float* __restrict__ atom,
                                              const int* __restrict__ adj,
                                              const float* __restrict__ W,
                                              const float* __restrict__ bias,
                                              float* __restrict__ out,
                                              float* __restrict__ ws) {
  __shared__ float As[128][36];            // 128 edges x 32 K, pad to 36 vs bank conflicts
  __shared__ float WkP[16 * WKP_STRIDE];   // 16 K-pairs x 128 cols x float2 (padded)
  __shared__ float sSum[C_];
  __shared__ float sSq[C_];

  const int b   = blockIdx.y;
  const int e0  = blockIdx.x * 128;
  const int tid = threadIdx.x;
  const int lane = tid & 31;
  const int wid  = tid >> 5;

  if (tid < C_) { sSum[tid] = 0.f; sSq[tid] = 0.f; }

  // Each pair of threads owns one edge for the A-tile gather.
  const int eL   = tid >> 1;      // 0..127
  const int half = tid & 1;
  const int* adjB = adj + ((size_t)b * E_ + e0) * 2;
  const int i0 = adjB[eL * 2 + 0];
  const int i1 = adjB[eL * 2 + 1];
  const float* atomB = atom + (size_t)b * N_ * F_;
  const float* rdB   = ws + WS_RD + b * C_;

  const v8f vzero = {0.f, 0.f, 0.f, 0.f, 0.f, 0.f, 0.f, 0.f};
  v8f acc[8];
#pragma unroll
  for (int n = 0; n < 8; ++n) acc[n] = vzero;

  for (int s = 0; s < 4; ++s) {
    __syncthreads();  // LDS reuse barrier (also publishes sSum/sSq init on s==0)

    // --- gather A slice: global channels [s*32, s*32+32) ---
    {
      const int src  = (s >= 2) ? i1 : i0;
      const int aoff = (s & 1) * 32 + half * 16;
      const float4* srcp = (const float4*)(atomB + src * F_ + aoff);
      float4* dstp = (float4*)(&As[eL][half * 16]);
#pragma unroll
      for (int j = 0; j < 4; ++j) dstp[j] = srcp[j];
    }
    // --- load W slice, scale by rdenom, interleave K-pairs ---
    // Task = (K-pair p in 0..15, col-group cg in 0..31); 512 tasks, 2/thread.
    {
#pragma unroll
      for (int j = 0; j < 2; ++j) {
        const int flat = tid + j * 256;   // 0..511
        const int p    = flat >> 5;       // 0..15
        const int cg   = flat & 31;       // float4 col group
        const int r0   = s * 32 + 2 * p;
        const float rd0 = rdB[r0];
        const float rd1 = rdB[r0 + 1];
        const float4 a4 = *(const float4*)(W + (size_t)r0 * C_ + cg * 4);
        const float4 b4 = *(const float4*)(W + (size_t)(r0 + 1) * C_ + cg * 4);
        float* dst = &WkP[p * WKP_STRIDE + cg * 8];
        const float4 lo = {a4.x * rd0, b4.x * rd1, a4.y * rd0, b4.y * rd1};
        const float4 hi = {a4.z * rd0, b4.z * rd1, a4.w * rd0, b4.w * rd1};
        *(float4*)(dst)     = lo;
        *(float4*)(dst + 4) = hi;
      }
    }
    __syncthreads();

    // --- WMMA: 8 K-steps of 16x16x4 f32, 8 N-tiles per wave ---
    const int arow = wid * 16 + (lane & 15);
    const int kk   = (lane >> 4) * 2;      // K sub-offset per A layout
    const int kp   = lane >> 4;            // which K-pair within the step
    const int bcol = lane & 15;
#pragma unroll
    for (int k = 0; k < 8; ++k) {
      v2f a;
      a.x = As[arow][k * 4 + kk];
      a.y = As[arow][k * 4 + kk + 1];
      const float* brow = &WkP[(2 * k + kp) * WKP_STRIDE];
#pragma unroll
      for (int n = 0; n < 8; ++n) {
        const v2f bb = *(const v2f*)(brow + (n * 16 + bcol) * 2);
        acc[n] = __builtin_amdgcn_wmma_f32_16x16x4_f32(
            false, a, false, bb, (short)0, acc[n], false, false);
      }
    }
  }

  // --- epilogue: +bias, store y, accumulate per-channel stats ---
  {
    const int col  = lane & 15;
    const int rsel = (lane >> 4) * 8;   // C/D layout: lanes>=16 hold M = 8+i
    float* outB = out + ((size_t)b * E_ + e0) * C_;
#pragma unroll
    for (int n = 0; n < 8; ++n) {
      const int ch = n * 16 + col;
      const float bs = bias[ch];
      float s = 0.f, q = 0.f;
#pragma unroll
      for (int i = 0; i < 8; ++i) {
        const float y = acc[n][i] + bs;
        const int m = wid * 16 + i + rsel;
        outB[(size_t)m * C_ + ch] = y;
        s += y;
        q += y * y;
      }
      atomicAdd(&sSum[ch], s);
      atomicAdd(&sSq[ch], q);
    }
  }
  __syncthreads();
  if (tid < C_) {
    atomicAdd(&ws[WS_SUM + tid], sSum[tid]);
    atomicAdd(&ws[WS_SQ  + tid], sSq[tid]);
  }
}

// ---------------------------------------------------------------------------
// Pass 3: finalize BN stats into fused scale/shift.
// ---------------------------------------------------------------------------
__global__ void k_stats(const float* __restrict__ gamma,
                        const float* __restrict__ beta,
                        float* __restrict__ ws) {
  const int c = threadIdx.x;
  if (c < C_) {
    const float inv  = 1.0f / (float)CNT_;
    const float mean = ws[WS_SUM + c] * inv;
    const float var  = ws[WS_SQ + c] * inv - mean * mean;   // biased variance
    const float rstd = rsqrtf(var + 1e-5f);
    const float sc   = gamma[c] * rstd;
    ws[WS_SCALE + c] = sc;
    ws[WS_SHIFT + c] = beta[c] - mean * sc;
  }
}

// ---------------------------------------------------------------------------
// Pass 4: out = tanh(y*scale + shift), float4-vectorized in place.
// ---------------------------------------------------------------------------
__global__ __launch_bounds__(256) void k_bn_tanh(float* __restrict__ out,
                                                 const float* __restrict__ ws) {
  const size_t v = (size_t)blockIdx.x * 256 + threadIdx.x;  // float4 index
  const int cg = (int)(v & 31);                             // channel group (C_/4 = 32)
  float4 y = ((float4*)out)[v];
  const float4 sc = *(const float4*)(ws + WS_SCALE + cg * 4);
  const float4 sh = *(const float4*)(ws + WS_SHIFT + cg * 4);
  y.x = tanhf(y.x * sc.x + sh.x);
  y.y = tanhf(y.y * sc.y + sh.y);
  y.z = tanhf(y.z * sc.z + sh.z);
  y.w = tanhf(y.w * sc.w + sh.w);
  ((float4*)out)[v] = y;
}

// ---------------------------------------------------------------------------
extern "C" void kernel_launch(void* const* d_in, const int* in_sizes, int n_in,
                              void* d_out, int out_size, void* d_ws, size_t ws_size,
                              hipStream_t stream) {
  const float* atom  = (const float*)d_in[0];   // (16, 2048, 64) f32
  const int*   adj   = (const int*)d_in[1];     // (16, 24576, 2) int
  const float* W     = (const float*)d_in[2];   // (128, 128) f32
  const float* bias  = (const float*)d_in[3];   // (128,) f32
  const float* gamma = (const float*)d_in[4];   // (128,) f32
  const float* beta  = (const float*)d_in[5];   // (128,) f32
  float* out = (float*)d_out;                   // (16, 24576, 128) f32
  float* ws  = (float*)d_ws;

  // Zero the accumulator scratch every call (harness poisons ws once, never re-poisons).
  hipMemsetAsync(ws, 0, WS_FLOATS * sizeof(float), stream);

  // 1) L1 sums over edges: 96 chunks of 256 edges per batch
  k_l1sum<<<dim3(96, B_), 256, 0, stream>>>(atom, adj, ws);
  // 1b) reciprocal denom with eps clamp (2048 values)
  k_rdenom<<<8, 256, 0, stream>>>(ws);
  // 2) WMMA GEMM + bias + stats accumulation: 192 edge-tiles x 16 batches
  k_gemm<<<dim3(E_ / 128, B_), 256, 0, stream>>>(atom, adj, W, bias, out, ws);
  // 3) BN stats -> scale/shift
  k_stats<<<1, 128, 0, stream>>>(gamma, beta, ws);
  // 4) BN + tanh elementwise (float4): 50331648/4 = 12582912 vec4 elements
  k_bn_tanh<<<(CNT_ * C_ / 4) / 256, 256, 0, stream>>>(out, ws);
}